// VectorizedConstrainedLoss_19181323944320
// MI455X (gfx1250) — compile-verified
//
#include <hip/hip_runtime.h>
#include <math.h>

// Problem constants (from reference setup_inputs)
#define MDIM 8
#define LDIM 11
#define NPART 512
#define HDIM 64
#define SNAPSHOTS (MDIM * LDIM)          // 88
#define NBATCH (MDIM * (LDIM - 1))       // 80
#define SIGMA_SQ 0.01f
#define LAMBDA_ANCHOR 1.0f
#define LAMBDA_DECAY 0.1f
#define R_DECAY 5.0f

typedef __attribute__((ext_vector_type(2))) float v2f;
typedef __attribute__((ext_vector_type(8))) float v8f;

#if __has_builtin(__builtin_amdgcn_wmma_f32_16x16x4_f32)
#define HAVE_WMMA_F32 1
#else
#define HAVE_WMMA_F32 0
#endif

__device__ __forceinline__ float fast_tanh(float x) {
#if __has_builtin(__builtin_amdgcn_tanhf)
  return __builtin_amdgcn_tanhf(x);
#elif __has_builtin(__builtin_amdgcn_tanh_f32)
  return __builtin_amdgcn_tanh_f32(x);
#else
  return tanhf(x);
#endif
}

// Raw hardware sqrt/rsq: skip libm's IEEE edge-case fixup chains
// (v_cmp_class / cndmask sequences) -- inputs here are clamped, well-scaled.
__device__ __forceinline__ float fast_sqrt(float x) {
#if __has_builtin(__builtin_amdgcn_sqrtf)
  return __builtin_amdgcn_sqrtf(x);
#else
  return sqrtf(x);
#endif
}

__device__ __forceinline__ float fast_rsq(float x) {
#if __has_builtin(__builtin_amdgcn_rsqf)
  return __builtin_amdgcn_rsqf(x);
#else
  return rsqrtf(x);
#endif
}

// ---------------------------------------------------------------------------
// Kernel A: pairwise pass. One wave per (snapshot, 16-row i tile).
// Per j: WMMA computes Z = r * P_W1 (16 pairs x 64 hidden, f32, K=0 only),
// tanh on TRANS pipe, per-lane accumulation of drift / laplacian / phi.
// Cross-lane reductions deferred to epilogue (shfl_xor).
// ---------------------------------------------------------------------------
__global__ void __launch_bounds__(32)
pairwise_kernel(const float* __restrict__ data,
                const float* __restrict__ P_W1, const float* __restrict__ P_b1,
                const float* __restrict__ P_W2,
                float* __restrict__ pairX, float* __restrict__ pairY,
                float* __restrict__ lapPhi, float* __restrict__ phiPart)
{
  const int s    = blockIdx.x;          // snapshot 0..87
  const int tile = blockIdx.y;          // 0..31
  const int lane = threadIdx.x;         // 0..31 (wave32)
  const int i0   = tile * 16;
  const int half = lane >> 4;
  const int hl   = lane & 15;
  const float* X = data + (size_t)s * NPART * 2;

  // Per-lane coefficients for h = ht*16 + (lane&15)
  float cw2[4], cww[4], ca[4], cb1[4], cw1[4];
  v2f Bm[4];
  #pragma unroll
  for (int ht = 0; ht < 4; ++ht) {
    const int h = ht * 16 + hl;
    const float w1 = P_W1[h], b1 = P_b1[h], w2 = P_W2[h];
    cw1[ht] = w1; cb1[ht] = b1; cw2[ht] = w2;
    cww[ht] = w1 * w2;
    ca[ht]  = w1 * w1 * w2;
    // B matrix (4x16): only K=0 row used -> (VGPR0, lanes 0-15)
    Bm[ht].x = (lane < 16) ? w1 : 0.0f;
    Bm[ht].y = 0.0f;
  }

  // Own row coords (A-matrix rows M=0..15 live in lanes 0..15)
  const float xox = X[(i0 + hl) * 2 + 0];
  const float xoy = X[(i0 + hl) * 2 + 1];

  // Per-VGPR-row broadcasts: row index i0+v+8*half for D-layout rows
  int   idxv[8];
  float xvx[8], xvy[8];
  #pragma unroll
  for (int v = 0; v < 8; ++v) {
    idxv[v] = i0 + v + 8 * half;
    xvx[v]  = X[idxv[v] * 2 + 0];
    xvy[v]  = X[idxv[v] * 2 + 1];
  }

  float dax[8], day[8], lac[8];
  #pragma unroll
  for (int v = 0; v < 8; ++v) { dax[v] = 0.f; day[v] = 0.f; lac[v] = 0.f; }
  float accphi[4] = {0.f, 0.f, 0.f, 0.f};   // split per-ht: shorter FMA chains

  for (int j = 0; j < NPART; ++j) {
    const float xjx = X[j * 2 + 0];
    const float xjy = X[j * 2 + 1];

    // A-matrix: rows = 16 i's, K=0 column = r_ij (raw distance)
    const float dox = xox - xjx, doy = xoy - xjy;
    const float r_own = fast_sqrt(dox * dox + doy * doy);
    v2f A;
    A.x = (lane < 16) ? r_own : 0.0f;   // K=0 (lanes 0-15) / K=2 (=0)
    A.y = 0.0f;                          // K=1 / K=3 unused

    v8f Z[4];
    #pragma unroll
    for (int ht = 0; ht < 4; ++ht) {
#if HAVE_WMMA_F32
      v8f c = {};
      Z[ht] = __builtin_amdgcn_wmma_f32_16x16x4_f32(
          false, A, false, Bm[ht], (short)0, c, false, false);
#else
      Z[ht] = (v8f){};
#endif
    }

    #pragma unroll
    for (int v = 0; v < 8; ++v) {
      const float dvx = xvx[v] - xjx;
      const float dvy = xvy[v] - xjy;
      const float rsq = dvx * dvx + dvy * dvy;
      const float inv = fast_rsq(fmaxf(rsq, 1e-20f));  // 1/max(r,1e-10)
      const float sel = (idxv[v] == j) ? 0.0f : 1.0f;  // mask diagonal
      float p1 = 0.f, p2 = 0.f;
      #pragma unroll
      for (int ht = 0; ht < 4; ++ht) {
#if HAVE_WMMA_F32
        const float z = Z[ht][v];
#else
        const float z = (rsq * inv) * cw1[ht];         // fallback: r*w1
#endif
        const float t    = fast_tanh(z + cb1[ht]);
        const float one  = fmaf(-t, t, 1.0f);          // 1 - t^2
        const float onem = one * sel;
        p1 = fmaf(cww[ht], onem, p1);                  // phi' partial
        p2 = fmaf(ca[ht], t * onem, p2);               // -phi''/2 partial
        accphi[ht] = fmaf(cw2[ht], t * sel, accphi[ht]); // phi partial
      }
      dax[v] = fmaf(p1, dvx * inv, dax[v]);            // g1 * unit_x
      day[v] = fmaf(p1, dvy * inv, day[v]);
      lac[v] = fmaf(p1, inv, lac[v]);                  // (d-1)/r * g1
      lac[v] = fmaf(-2.0f, p2, lac[v]);                // + phi''
    }
  }

  // Epilogue: reduce over the 16 lanes of each half (h components)
  #pragma unroll
  for (int v = 0; v < 8; ++v) {
    float sx = dax[v], sy = day[v], sl = lac[v];
    #pragma unroll
    for (int m = 1; m <= 8; m <<= 1) {
      sx += __shfl_xor(sx, m, 32);
      sy += __shfl_xor(sy, m, 32);
      sl += __shfl_xor(sl, m, 32);
    }
    if (hl == 0) {  // lane 0 writes row i0+v, lane 16 writes row i0+v+8
      const int gi = s * NPART + idxv[v];
      pairX[gi]  = sx;
      pairY[gi]  = sy;
      lapPhi[gi] = sl;
    }
  }
  float ap = (accphi[0] + accphi[1]) + (accphi[2] + accphi[3]);
  #pragma unroll
  for (int m = 1; m <= 16; m <<= 1) ap += __shfl_xor(ap, m, 32);
  if (lane == 0) phiPart[s * 32 + tile] = ap;
}

// ---------------------------------------------------------------------------
// Kernel B: per-point MLP (grad V, lap V, V). 512x smaller than kernel A;
// scalar-uniform weight loads, one point per thread.
// ---------------------------------------------------------------------------
__global__ void __launch_bounds__(256)
point_kernel(const float* __restrict__ data,
             const float* __restrict__ V_W1, const float* __restrict__ V_b1,
             const float* __restrict__ V_W2, const float* __restrict__ V_b2,
             float* __restrict__ gVx, float* __restrict__ gVy,
             float* __restrict__ lapV, float* __restrict__ Vval)
{
  const int p = blockIdx.x * blockDim.x + threadIdx.x;
  if (p >= SNAPSHOTS * NPART) return;
  const float x = data[p * 2 + 0];
  const float y = data[p * 2 + 1];
  float gv = 0.f, gx = 0.f, gy = 0.f, lp = 0.f;
  for (int h = 0; h < HDIM; ++h) {
    const float w1x = V_W1[h];          // V_W1 is (d,H) row-major
    const float w1y = V_W1[HDIM + h];
    const float b1  = V_b1[h];
    const float w2  = V_W2[h];
    const float t   = fast_tanh(fmaf(x, w1x, fmaf(y, w1y, b1)));
    const float one = fmaf(-t, t, 1.0f);
    gv = fmaf(w2, t, gv);
    gx = fmaf(w1x * w2, one, gx);
    gy = fmaf(w1y * w2, one, gy);
    lp = fmaf((w1x * w1x + w1y * w1y) * w2, t * one, lp);
  }
  gVx[p]  = gx;
  gVy[p]  = gy;
  lapV[p] = -2.0f * lp;
  Vval[p] = gv + V_b2[0];
}

// ---------------------------------------------------------------------------
// Kernel C: final reduction -> scalar loss. Single block.
// Energy telescopes: sum_b (E_next - E_curr) = sum_m (E[m,L-1] - E[m,0]);
// the per-pair +P_b2 constant cancels exactly in this weighted sum.
// ---------------------------------------------------------------------------
__global__ void __launch_bounds__(256)
reduce_kernel(const float* __restrict__ tsnap,
              const float* __restrict__ V_b1, const float* __restrict__ V_W2,
              const float* __restrict__ V_b2,
              const float* __restrict__ P_W1, const float* __restrict__ P_b1,
              const float* __restrict__ P_W2, const float* __restrict__ P_b2,
              const float* __restrict__ pairX, const float* __restrict__ pairY,
              const float* __restrict__ lapPhi,
              const float* __restrict__ gVx, const float* __restrict__ gVy,
              const float* __restrict__ lapV, const float* __restrict__ Vval,
              const float* __restrict__ phiPart,
              float* __restrict__ out)
{
  const int tid = threadIdx.x;
  const float invN = 1.0f / (float)NPART;
  float acc = 0.f;

  // J_diss + J_diff over b = m*10 + l (l=0..9), snapshot s = m*11 + l
  for (int idx = tid; idx < NBATCH * NPART; idx += 256) {
    const int b = idx >> 9;
    const int i = idx & (NPART - 1);
    const int m = b / 10;
    const int l = b - m * 10;
    const int g = (m * LDIM + l) * NPART + i;
    const float dt  = tsnap[l + 1] - tsnap[l];
    const float dx  = -gVx[g] - pairX[g] * invN;
    const float dy  = -gVy[g] - pairY[g] * invN;
    const float lap = lapV[g] + lapPhi[g] * invN;
    acc += dt * invN * (dx * dx + dy * dy + SIGMA_SQ * lap);
  }

  // -2 * energy term, V part: only l=0 (w=-1) and l=L-1 (w=+1) survive
  for (int idx = tid; idx < 2 * MDIM * NPART; idx += 256) {
    const int k = idx >> 9;
    const int i = idx & (NPART - 1);
    const int m = k >> 1;
    const int l = (k & 1) ? (LDIM - 1) : 0;
    const float w = (k & 1) ? 1.0f : -1.0f;
    acc += -2.0f * w * Vval[(m * LDIM + l) * NPART + i] * invN;
  }
  // -2 * energy term, pairwise phi part
  for (int idx = tid; idx < 2 * MDIM * 32; idx += 256) {
    const int k = idx >> 5;
    const int t32 = idx & 31;
    const int m = k >> 1;
    const int l = (k & 1) ? (LDIM - 1) : 0;
    const float w = (k & 1) ? 1.0f : -1.0f;
    acc += -2.0f * w * phiPart[(m * LDIM + l) * 32 + t32] * (invN * invN);
  }

  __shared__ float red[256];
  red[tid] = acc;
  __syncthreads();
  for (int st = 128; st > 0; st >>= 1) {
    if (tid < st) red[tid] += red[tid + st];
    __syncthreads();
  }
  if (tid == 0) {
    const float residual = red[0] / (float)NBATCH;
    float loss = residual * residual;
    float v0 = 0.f, p5 = 0.f;
    for (int h = 0; h < HDIM; ++h) {
      v0 = fmaf(V_W2[h], fast_tanh(V_b1[h]), v0);
      p5 = fmaf(P_W2[h], fast_tanh(fmaf(R_DECAY, P_W1[h], P_b1[h])), p5);
    }
    v0 += V_b2[0];
    p5 += P_b2[0];
    out[0] = loss + LAMBDA_ANCHOR * v0 * v0 + LAMBDA_DECAY * p5 * p5;
  }
}

// ---------------------------------------------------------------------------
extern "C" void kernel_launch(void* const* d_in, const int* in_sizes, int n_in,
                              void* d_out, int out_size, void* d_ws, size_t ws_size,
                              hipStream_t stream) {
  const float* data = (const float*)d_in[0];
  const float* tsn  = (const float*)d_in[1];
  const float* V_W1 = (const float*)d_in[2];
  const float* V_b1 = (const float*)d_in[3];
  const float* V_W2 = (const float*)d_in[4];
  const float* V_b2 = (const float*)d_in[5];
  const float* P_W1 = (const float*)d_in[6];
  const float* P_b1 = (const float*)d_in[7];
  const float* P_W2 = (const float*)d_in[8];
  const float* P_b2 = (const float*)d_in[9];

  float* ws = (float*)d_ws;
  const int NP = SNAPSHOTS * NPART;   // 45056
  float* pairX   = ws + 0 * NP;
  float* pairY   = ws + 1 * NP;
  float* lapPhi  = ws + 2 * NP;
  float* gVx     = ws + 3 * NP;
  float* gVy     = ws + 4 * NP;
  float* lapV    = ws + 5 * NP;
  float* Vval    = ws + 6 * NP;
  float* phiPart = ws + 7 * NP;       // 88*32 = 2816 floats

  pairwise_kernel<<<dim3(SNAPSHOTS, NPART / 16), 32, 0, stream>>>(
      data, P_W1, P_b1, P_W2, pairX, pairY, lapPhi, phiPart);

  point_kernel<<<(NP + 255) / 256, 256, 0, stream>>>(
      data, V_W1, V_b1, V_W2, V_b2, gVx, gVy, lapV, Vval);

  reduce_kernel<<<1, 256, 0, stream>>>(
      tsn, V_b1, V_W2, V_b2, P_W1, P_b1, P_W2, P_b2,
      pairX, pairY, lapPhi, gVx, gVy, lapV, Vval, phiPart,
      (float*)d_out);
}